// L_spa_89498528514824
// MI455X (gfx1250) — compile-verified
//
#include <hip/hip_runtime.h>

// MI455X / gfx1250: memory-bound streaming reduction.
// 691 MB read -> 1 float.  Roofline floor ~30us @ 23.3 TB/s.

typedef float v2f __attribute__((ext_vector_type(2)));
typedef float v8f __attribute__((ext_vector_type(8)));

#define WIN   12
#define BLOCK 256   // 8 wave32s

// Deterministic block reduction of BLOCK partials using chained
// V_WMMA_F32_16X16X4_F32 accumulation (B = ones => D[m,n] = sum_k A[m,k] + C).
// A vgpr0 holds one value per lane: A[m,0]=v_m (lanes 0-15), A[m,2]=v_{m+16}
// (lanes 16-31); vgpr1 = 0.  After 8 chained WMMAs, total = sum over the 16
// row-sums, read from D vgpr r at lanes 0 (M=r) and 16 (M=r+8).
__device__ __forceinline__ float block_reduce_wmma(float partial, float* lds) {
    lds[threadIdx.x] = partial;
    __syncthreads();
    float tot = 0.0f;
    if (threadIdx.x < 32) {              // wave 0 only; EXEC all-ones inside
        v8f acc = {};
        v2f b;  b.x = 1.0f; b.y = 1.0f;  // B[k,n] = 1 for all k,n
        #pragma unroll
        for (int g = 0; g < BLOCK / 32; ++g) {
            v2f a;
            a.x = lds[g * 32 + threadIdx.x];
            a.y = 0.0f;
            acc = __builtin_amdgcn_wmma_f32_16x16x4_f32(
                false, a, false, b, (short)0, acc, false, false);
        }
        #pragma unroll
        for (int r = 0; r < 8; ++r)
            tot += __shfl(acc[r], 0, 32) + __shfl(acc[r], 16, 32);
    }
    return tot;  // valid in lane 0 of wave 0
}

// One window per lane.  Window n, pixel (h,w): element at (h*12+w)*N + n,
// 3 floats.  Per-pixel base is wave-uniform (SGPR); n*3 is the VGPR offset.
__global__ void __launch_bounds__(BLOCK)
spa_partial(const float* __restrict__ s1, const float* __restrict__ s2,
            float* __restrict__ ws, int N) {
    __shared__ float lds[BLOCK];
    const int n = blockIdx.x * BLOCK + threadIdx.x;

    float q[9];
    #pragma unroll
    for (int i = 0; i < 9; ++i) q[i] = 0.0f;

    if (n < N) {
        const size_t pixStride = (size_t)N * 3;   // floats between pixels
        const size_t off       = (size_t)n * 3;   // per-lane offset (fits u32*4)
        #pragma unroll
        for (int h = 0; h < WIN; ++h) {
            #pragma unroll
            for (int w = 0; w < WIN; ++w) {
                const int pix = h * WIN + w;
                const float* pa = s1 + (size_t)pix * pixStride + off;
                const float* pb = s2 + (size_t)pix * pixStride + off;
                // streaming: read-once -> non-temporal; vectorizer should
                // merge each triple into global_load_b96 ... nt
                float a0 = __builtin_nontemporal_load(pa + 0);
                float a1 = __builtin_nontemporal_load(pa + 1);
                float a2 = __builtin_nontemporal_load(pa + 2);
                float b0 = __builtin_nontemporal_load(pb + 0);
                float b1 = __builtin_nontemporal_load(pb + 1);
                float b2 = __builtin_nontemporal_load(pb + 2);
                const int bi = (h >> 2) * 3 + (w >> 2);  // compile-time index
                q[bi] += (a0 + a1 + a2) - (b0 + b1 + b2);
            }
        }
    }

    // E[n]: padded 4-direction gradient energy of q (q_true = q/48, deferred).
    // left+right over a row (r0,r1,r2): r0^2 + r2^2 + 2(r0-r1)^2 + 2(r1-r2)^2
    // up+down over a column: same form.
    float e = 0.0f;
    #pragma unroll
    for (int y = 0; y < 3; ++y) {
        float r0 = q[y * 3 + 0], r1 = q[y * 3 + 1], r2 = q[y * 3 + 2];
        float dr01 = r0 - r1, dr12 = r1 - r2;
        e += r0 * r0 + r2 * r2 + 2.0f * (dr01 * dr01 + dr12 * dr12);
        float c0 = q[y], c1 = q[3 + y], c2 = q[6 + y];
        float dc01 = c0 - c1, dc12 = c1 - c2;
        e += c0 * c0 + c2 * c2 + 2.0f * (dc01 * dc01 + dc12 * dc12);
    }
    // lanes with n >= N have q == 0 -> e == 0, and still join the reduction
    // (WMMA requires EXEC all-ones; no early return).

    float tot = block_reduce_wmma(e, lds);
    if (threadIdx.x == 0) ws[blockIdx.x] = tot;
}

__global__ void __launch_bounds__(BLOCK)
spa_final(const float* __restrict__ ws, float* __restrict__ out,
          int numBlocks, float scale) {
    __shared__ float lds[BLOCK];
    float s = 0.0f;
    for (int i = threadIdx.x; i < numBlocks; i += BLOCK) s += ws[i];
    float tot = block_reduce_wmma(s, lds);
    if (threadIdx.x == 0) out[0] = tot * scale;
}

extern "C" void kernel_launch(void* const* d_in, const int* in_sizes, int n_in,
                              void* d_out, int out_size, void* d_ws, size_t ws_size,
                              hipStream_t stream) {
    const float* s1 = (const float*)d_in[0];
    const float* s2 = (const float*)d_in[1];
    // in_sizes[0] = 12*12*N*3  -> recover N on host (no device reads in capture)
    const int N = in_sizes[0] / (WIN * WIN * 3);
    const int blocks = (N + BLOCK - 1) / BLOCK;   // ~782 blocks; ws needs 4*blocks bytes

    float* ws = (float*)d_ws;
    spa_partial<<<blocks, BLOCK, 0, stream>>>(s1, s2, ws, N);

    // q_true = q/48 -> E scales by 1/48^2; mean over N*9 elements.
    const float scale = (float)(1.0 / (48.0 * 48.0 * 9.0 * (double)N));
    spa_final<<<1, BLOCK, 0, stream>>>(ws, (float*)d_out, blocks, scale);
}